// GnnSynthetic_76639396430551
// MI455X (gfx1250) — compile-verified
//
#include <hip/hip_runtime.h>
#include <hip/hip_bf16.h>

typedef float v8f __attribute__((ext_vector_type(8)));
typedef float v2f __attribute__((ext_vector_type(2)));

#define NFEAT 64
#define NEMB  192
#define MAXBLKS 128

// ---------------- fill ----------------
__global__ void fill_f32(float* __restrict__ p, float v, long n) {
    long i = (long)blockIdx.x * blockDim.x + threadIdx.x;
    if (i < n) p[i] = v;
}

// ---------------- degree count (self-loop pre-seeded with 1.0) ----------------
__global__ void deg_count(const int* __restrict__ dst, float* __restrict__ deg, int E) {
    int stride = gridDim.x * blockDim.x;
    for (int e = blockIdx.x * blockDim.x + threadIdx.x; e < E; e += stride)
        atomicAdd(&deg[dst[e]], 1.0f);
}

__global__ void make_dinv(float* __restrict__ deg, int n) {
    int i = blockIdx.x * blockDim.x + threadIdx.x;
    if (i < n) deg[i] = rsqrtf(deg[i]);   // deg >= 1 always (self loop)
}

// ---------------- GEMM: H[n,64] = A[n,lda-strided] @ W[64,64], via f32 WMMA ----------------
// One wave computes one 16x16 tile of H. 16 chained V_WMMA_F32_16X16X4_F32 over K=64.
__global__ void gemm_n64(const float* __restrict__ A, int lda,
                         const float* __restrict__ W,
                         float* __restrict__ H, int n) {
    int wave = (int)((blockIdx.x * blockDim.x + threadIdx.x) >> 5);
    int lane = threadIdx.x & 31;
    int tileM = (wave >> 2) * 16;        // 64/16 = 4 tiles along N-dim of output
    int tileN = (wave & 3) * 16;
    if (tileM >= n) return;              // wave-uniform: EXEC stays all-ones for WMMA

    int hi = lane >> 4;                  // 0: K-pair {0,1}, 1: K-pair {2,3}
    int lo = lane & 15;

    v8f c = {};
    const float* arow = A + (long)(tileM + lo) * lda + 2 * hi;   // A(m, k)
    const float* bcol = W + (long)(2 * hi) * NFEAT + tileN + lo; // W(k, n)
    #pragma unroll
    for (int k = 0; k < NFEAT; k += 4) {
        v2f a, b;
        a.x = arow[0];        a.y = arow[1];          // K = k+2hi, k+2hi+1
        b.x = bcol[0];        b.y = bcol[NFEAT];      // same K rows of W
        c = __builtin_amdgcn_wmma_f32_16x16x4_f32(
                /*neg_a=*/false, a, /*neg_b=*/false, b,
                /*c_mod=*/(short)0, c, /*reuse_a=*/false, /*reuse_b=*/false);
        arow += 4;
        bcol += 4 * NFEAT;
    }
    // D layout: VGPR r -> row tileM + r + 8*hi, col tileN + lo
    float* out = H + (long)(tileM + 8 * hi) * NFEAT + tileN + lo;
    #pragma unroll
    for (int r = 0; r < 8; ++r) out[(long)r * NFEAT] = c[r];
}

// ---------------- edge scatter: agg[dst] += h[src] * dinv[src]*dinv[dst] ----------------
// One wave per edge; lane l handles features l and l+32 (coalesced 256B row).
__global__ void scatter_edges(const int* __restrict__ src, const int* __restrict__ dst,
                              const float* __restrict__ dinv,
                              const float* __restrict__ h,
                              float* __restrict__ agg, int E) {
    int lane = threadIdx.x & 31;
    int wave = (int)((blockIdx.x * blockDim.x + threadIdx.x) >> 5);
    int nwaves = (gridDim.x * blockDim.x) >> 5;
    for (int e = wave; e < E; e += nwaves) {
        int s = src[e];
        int d = dst[e];
        float norm = dinv[s] * dinv[d];
        const float* hs = h + (long)s * NFEAT;
        float*       ad = agg + (long)d * NFEAT;
        float v0 = hs[lane]      * norm;
        float v1 = hs[lane + 32] * norm;
        atomicAdd(&ad[lane],      v0);
        atomicAdd(&ad[lane + 32], v1);
    }
}

// ---------------- epilogue: emb[i, colOff+f] = act(agg + h*dinv^2 + b) ----------------
__global__ void finish_layer(const float* __restrict__ agg, const float* __restrict__ h,
                             const float* __restrict__ dinv, const float* __restrict__ b,
                             float* __restrict__ emb, int colOff, int relu, int n) {
    long idx = (long)blockIdx.x * blockDim.x + threadIdx.x;
    if (idx >= (long)n * NFEAT) return;
    int i = (int)(idx >> 6);
    int f = (int)(idx & 63);
    float di = dinv[i];
    float v = agg[idx] + h[idx] * di * di + b[f];   // self-loop folded in here
    if (relu) v = fmaxf(v, 0.0f);
    emb[(long)i * NEMB + colOff + f] = v;
}

// ---------------- graph max, stage 1: per-block partial max over a row slab ----------------
__global__ void max_stage1(const float* __restrict__ emb, float* __restrict__ partial, int n) {
    int f = threadIdx.x;                 // 192 threads: feature id (coalesced within row)
    int rowsPer = (n + gridDim.x - 1) / gridDim.x;
    int r0 = blockIdx.x * rowsPer;
    int r1 = min(n, r0 + rowsPer);
    float m = -INFINITY;
    for (int r = r0; r < r1; ++r)
        m = fmaxf(m, emb[(long)r * NEMB + f]);
    partial[(long)blockIdx.x * NEMB + f] = m;
}

__global__ void max_stage2(const float* __restrict__ partial, float* __restrict__ ge, int nb) {
    int f = threadIdx.x;
    float m = -INFINITY;
    for (int b = 0; b < nb; ++b)
        m = fmaxf(m, partial[(long)b * NEMB + f]);
    ge[f] = m;
}

// ---------------- target-node logits: out[c] = emb[t,:] @ fcW[:,c] + fcb[c] ----------------
__global__ void target_logits(const float* __restrict__ emb, const float* __restrict__ fcW,
                              const float* __restrict__ fcb, const int* __restrict__ target,
                              float* __restrict__ out) {
    int c = threadIdx.x;
    if (c >= 4) return;
    int t = target[0];
    float s = fcb[c];
    const float* row = emb + (long)t * NEMB;
    for (int f = 0; f < NEMB; ++f)
        s += row[f] * fcW[f * 4 + c];
    out[c] = s;
}

extern "C" void kernel_launch(void* const* d_in, const int* in_sizes, int n_in,
                              void* d_out, int out_size, void* d_ws, size_t ws_size,
                              hipStream_t stream) {
    const float* x      = (const float*)d_in[0];
    const int*   edge   = (const int*)d_in[1];   // [2, E]: src row then dst row
    const int*   target = (const int*)d_in[3];
    const float* W1 = (const float*)d_in[4];
    const float* b1 = (const float*)d_in[5];
    const float* W2 = (const float*)d_in[6];
    const float* b2 = (const float*)d_in[7];
    const float* W3 = (const float*)d_in[8];
    const float* b3 = (const float*)d_in[9];
    const float* fcW = (const float*)d_in[10];
    const float* fcb = (const float*)d_in[11];

    const int N = in_sizes[0] / NFEAT;
    const int E = in_sizes[1] / 2;
    const int* esrc = edge;
    const int* edst = edge + E;

    // workspace layout (floats)
    float* ws   = (float*)d_ws;
    float* dinv = ws;                             // N   (degree, then rsqrt in-place)
    float* h    = dinv + N;                       // N*64
    float* agg  = h + (long)N * NFEAT;            // N*64
    float* part = agg + (long)N * NFEAT;          // MAXBLKS*192

    // output layout: emb [N,192] | graph_embedding [192] | logits [4]
    float* emb = (float*)d_out;
    float* ge  = emb + (long)N * NEMB;
    float* out4 = ge + NEMB;

    const long nh = (long)N * NFEAT;
    const int blkFill = (int)((nh + 255) / 256);

    // ---- GCN normalization: deg = 1 (self loop) + in-degree; dinv = rsqrt(deg) ----
    fill_f32<<<(N + 255) / 256, 256, 0, stream>>>(dinv, 1.0f, N);
    deg_count<<<1024, 256, 0, stream>>>(edst, dinv, E);
    make_dinv<<<(N + 255) / 256, 256, 0, stream>>>(dinv, N);

    // ---- three GCN layers; N = 3125*16 so tiles divide exactly ----
    const int gemmWaves  = (N / 16) * 4;
    const int gemmBlocks = (gemmWaves + 7) / 8;   // 8 waves (256 threads) per block

    struct { const float* A; int lda; const float* W; const float* b; int colOff; int relu; }
    layers[3] = {
        { x,        NFEAT, W1, b1, 0,   1 },
        { emb + 0,  NEMB,  W2, b2, 64,  1 },
        { emb + 64, NEMB,  W3, b3, 128, 0 },
    };

    for (int l = 0; l < 3; ++l) {
        gemm_n64<<<gemmBlocks, 256, 0, stream>>>(layers[l].A, layers[l].lda, layers[l].W, h, N);
        fill_f32<<<blkFill, 256, 0, stream>>>(agg, 0.0f, nh);
        scatter_edges<<<2048, 256, 0, stream>>>(esrc, edst, dinv, h, agg, E);
        finish_layer<<<blkFill, 256, 0, stream>>>(agg, h, dinv, layers[l].b,
                                                  emb, layers[l].colOff, layers[l].relu, N);
    }

    // ---- graph embedding (global max over nodes) ----
    max_stage1<<<MAXBLKS, NEMB, 0, stream>>>(emb, part, N);
    max_stage2<<<1, NEMB, 0, stream>>>(part, ge, MAXBLKS);

    // ---- target node logits ----
    target_logits<<<1, 32, 0, stream>>>(emb, fcW, fcb, target, out4);
}